// ScatterRouter_4054449127994
// MI455X (gfx1250) — compile-verified
//
#include <hip/hip_runtime.h>
#include <hip/hip_bf16.h>
#include <cstdint>

#define N_TOK   8192
#define DIM     1024
#define PATHS   8

typedef float f32x4 __attribute__((ext_vector_type(4)));

// ---------------------------------------------------------------------------
// Kernel A: top-2 gating + stable compaction offsets (single workgroup scan).
//   pos_ws [N][P]  : destination row of token n inside path p's buffer
//   gate_ws[N][P]  : score if selected, 0 otherwise
//   tags   [P][N]  : original token index for each output row (as float)
//   loads  [P]     : number of selected tokens per path (as float)
// ---------------------------------------------------------------------------
__global__ __launch_bounds__(1024)
void router_topk_scan(const float* __restrict__ score,
                      int*   __restrict__ pos_ws,
                      float* __restrict__ gate_ws,
                      float* __restrict__ tags_out,
                      float* __restrict__ loads_out) {
    const int t = threadIdx.x;                 // 0..1023, 8 tokens per thread
    __shared__ int sc[PATHS][1024];

    unsigned msk[8];
    int cnt[PATHS];
#pragma unroll
    for (int p = 0; p < PATHS; ++p) cnt[p] = 0;

    // ---- pass 1: top-2 per token (jax.lax.top_k tie-break: lowest index) ----
#pragma unroll
    for (int i = 0; i < 8; ++i) {
        const int n = t * 8 + i;
        const f32x4* s4 = (const f32x4*)(score + (size_t)n * PATHS);
        f32x4 a = s4[0], b = s4[1];
        float s[PATHS] = {a.x, a.y, a.z, a.w, b.x, b.y, b.z, b.w};

        int i1 = 0; float v1 = s[0];
#pragma unroll
        for (int p = 1; p < PATHS; ++p)
            if (s[p] > v1) { v1 = s[p]; i1 = p; }
        int i2 = -1; float v2 = -__builtin_inff();
#pragma unroll
        for (int p = 0; p < PATHS; ++p)
            if (p != i1 && s[p] > v2) { v2 = s[p]; i2 = p; }

        const unsigned m = (1u << i1) | (1u << i2);
        msk[i] = m;
        cnt[i1]++; cnt[i2]++;

        float g[PATHS];
#pragma unroll
        for (int p = 0; p < PATHS; ++p)
            g[p] = ((m >> p) & 1u) ? s[p] : 0.0f;
        f32x4* g4 = (f32x4*)(gate_ws + (size_t)n * PATHS);
        g4[0] = (f32x4){g[0], g[1], g[2], g[3]};
        g4[1] = (f32x4){g[4], g[5], g[6], g[7]};
    }

    // ---- block-wide inclusive scan over thread-local counts, all 8 paths ----
#pragma unroll
    for (int p = 0; p < PATHS; ++p) sc[p][t] = cnt[p];
    __syncthreads();
    for (int off = 1; off < 1024; off <<= 1) {
        int v[PATHS];
#pragma unroll
        for (int p = 0; p < PATHS; ++p) v[p] = (t >= off) ? sc[p][t - off] : 0;
        __syncthreads();
#pragma unroll
        for (int p = 0; p < PATHS; ++p) sc[p][t] += v[p];
        __syncthreads();
    }

    int load_p[PATHS], run[PATHS];
#pragma unroll
    for (int p = 0; p < PATHS; ++p) {
        load_p[p] = sc[p][1023];
        run[p]    = sc[p][t] - cnt[p];   // selected tokens before this thread
    }
    if (t == 0) {
#pragma unroll
        for (int p = 0; p < PATHS; ++p) loads_out[p] = (float)load_p[p];
    }

    // ---- pass 2: emit destination rows + tags (bijection per path) ----------
#pragma unroll
    for (int i = 0; i < 8; ++i) {
        const int n = t * 8 + i;
        const unsigned m = msk[i];
#pragma unroll
        for (int p = 0; p < PATHS; ++p) {
            int pp;
            if ((m >> p) & 1u) { pp = run[p]; run[p]++; }
            else               { pp = load_p[p] + (n - run[p]); }
            pos_ws[(size_t)n * PATHS + p] = pp;
            tags_out[(size_t)p * N_TOK + pp] = (float)n;
        }
    }
}

// ---------------------------------------------------------------------------
// Kernel B: stream one in_flow row per block through LDS (async load, NT),
// emit 8 output rows: 2 scaled (ds_load_b128 -> v_mul -> NT b128 store),
// 6 zero rows pushed with global_store_async_from_lds_b128 (NT) from a
// zeroed LDS region — no VALU/VGPR traffic for 192 MB of the 256 MB output.
// ---------------------------------------------------------------------------
__global__ __launch_bounds__(256)
void router_dispatch(const float* __restrict__ in_flow,
                     const int*   __restrict__ pos_ws,
                     const float* __restrict__ gate_ws,
                     float* __restrict__ out_flows) {
    const int n = blockIdx.x;          // token
    const int t = threadIdx.x;         // 0..255, 4 floats (16B) each

    __shared__ float row[DIM];
    __shared__ float zeros[DIM];

    // zero region (one b128 store per lane), reused for all padded rows
    *(f32x4*)&zeros[t * 4] = (f32x4){0.f, 0.f, 0.f, 0.f};

    // async stage of the 4KB token row: 256 lanes x 16B = whole row.
    // Single-use data -> non-temporal load policy.
    {
        unsigned long long gsrc =
            (unsigned long long)(const void*)(in_flow + (size_t)n * DIM + t * 4);
        unsigned ldst = (unsigned)(uintptr_t)(void*)&row[t * 4];
        asm volatile("global_load_async_to_lds_b128 %0, %1, off th:TH_LOAD_NT"
                     :: "v"(ldst), "v"(gsrc) : "memory");
    }
    asm volatile("s_wait_asynccnt 0x0" ::: "memory");
    __syncthreads();   // zeros[] visible + every wave's async chunk landed

#pragma unroll
    for (int p = 0; p < PATHS; ++p) {
        const float scale = gate_ws[(size_t)n * PATHS + p];   // uniform
        const int   r     = pos_ws[(size_t)n * PATHS + p];    // uniform
        float* dst = out_flows + ((size_t)p * N_TOK + (size_t)r) * DIM + t * 4;

        if (scale != 0.0f) {
            f32x4 v = *(const f32x4*)&row[t * 4];
            v *= scale;
            __builtin_nontemporal_store(v, (f32x4*)dst);     // b128 th:NT
        } else {
            unsigned long long gdst = (unsigned long long)(void*)dst;
            unsigned lsrc = (unsigned)(uintptr_t)(void*)&zeros[t * 4];
            asm volatile("global_store_async_from_lds_b128 %0, %1, off th:TH_STORE_NT"
                         :: "v"(gdst), "v"(lsrc) : "memory");
        }
    }
    asm volatile("s_wait_asynccnt 0x0" ::: "memory");
}

// ---------------------------------------------------------------------------
extern "C" void kernel_launch(void* const* d_in, const int* in_sizes, int n_in,
                              void* d_out, int out_size, void* d_ws, size_t ws_size,
                              hipStream_t stream) {
    const float* in_flow = (const float*)d_in[0];   // [8192,1024] f32
    const float* score   = (const float*)d_in[1];   // [8192,8]    f32

    float* out_flows = (float*)d_out;                                   // [8,8192,1024]
    float* tags      = out_flows + (size_t)PATHS * N_TOK * DIM;         // [8,8192]
    float* loads     = tags      + (size_t)PATHS * N_TOK;               // [8]

    int*   pos_ws  = (int*)d_ws;                                        // 256 KB
    float* gate_ws = (float*)((char*)d_ws + (size_t)N_TOK * PATHS * 4); // 256 KB

    router_topk_scan<<<1, 1024, 0, stream>>>(score, pos_ws, gate_ws, tags, loads);
    router_dispatch<<<N_TOK, 256, 0, stream>>>(in_flow, pos_ws, gate_ws, out_flows);
}